// RMSELoss_43293270343860
// MI455X (gfx1250) — compile-verified
//
#include <hip/hip_runtime.h>
#include <math.h>

#define NBLOCKS  2048
#define NTHREADS 256
#define NWAVES   (NTHREADS / 32)

typedef __attribute__((ext_vector_type(2))) float v2f;
typedef __attribute__((ext_vector_type(8))) float v8f;

// weight = 1.5 if diff>=3, 2.0 if diff<0, else 1.0 ; accumulate (diff*w)^2
__device__ __forceinline__ float weighted_sq(float x, float t) {
    float d = t - x;
    float w = (d >= 3.0f) ? 1.5f : ((d < 0.0f) ? 2.0f : 1.0f);
    float s = d * w;
    return s * s;
}

// Pass 1: grid-stride streaming over float4 (global_load_b128), per-block partial.
__global__ void rmse_partial_kernel(const float* __restrict__ in,
                                    const float* __restrict__ tgt,
                                    float* __restrict__ partial,
                                    long long n) {
    const long long n4     = n >> 2;
    const long long tid    = (long long)blockIdx.x * blockDim.x + threadIdx.x;
    const long long stride = (long long)gridDim.x * blockDim.x;
    const float4* __restrict__ in4 = (const float4*)in;
    const float4* __restrict__ tg4 = (const float4*)tgt;

    float acc = 0.0f;
    for (long long i = tid; i < n4; i += stride) {
        float4 a = in4[i];
        float4 b = tg4[i];
        acc += weighted_sq(a.x, b.x);
        acc += weighted_sq(a.y, b.y);
        acc += weighted_sq(a.z, b.z);
        acc += weighted_sq(a.w, b.w);
    }
    // tail (N % 4 != 0 handled generically; first few global threads take it)
    const long long base = n4 << 2;
    if (tid < (n - base)) {
        acc += weighted_sq(in[base + tid], tgt[base + tid]);
    }

    // wave32 shuffle tree
    #pragma unroll
    for (int o = 16; o > 0; o >>= 1)
        acc += __shfl_xor(acc, o, 32);

    __shared__ float lds[NWAVES];
    const int lane = threadIdx.x & 31;
    const int wave = threadIdx.x >> 5;
    if (lane == 0) lds[wave] = acc;
    __syncthreads();

    if (wave == 0) {
        float v = (lane < NWAVES) ? lds[lane] : 0.0f;
        #pragma unroll
        for (int o = NWAVES >> 1; o > 0; o >>= 1)
            v += __shfl_xor(v, o, 32);
        if (lane == 0) partial[blockIdx.x] = v;
    }
}

// Pass 2: single block reduces NBLOCKS partials; final 32->1 uses
// V_WMMA_F32_16X16X4_F32 with B = ones (D rows become row-sums of A).
__global__ void rmse_final_kernel(const float* __restrict__ partial,
                                  float* __restrict__ out,
                                  long long n) {
    __shared__ float lds[NTHREADS];
    float s = 0.0f;
    for (int i = threadIdx.x; i < NBLOCKS; i += NTHREADS)
        s += partial[i];
    lds[threadIdx.x] = s;
    __syncthreads();

    if (threadIdx.x < 32) {          // whole wave 0, EXEC all-ones inside
        float v = 0.0f;
        #pragma unroll
        for (int j = 0; j < NTHREADS / 32; ++j)
            v += lds[threadIdx.x + 32 * j];

        // A (16x4 f32): lane L, VGPR0 -> (M=L%16, K = L<16 ? 0 : 2); VGPR1 zeroed.
        // B (4x16) all ones  =>  D[M][N] = v(lane M) + v(lane M+16), any N.
        v2f a; a[0] = v;    a[1] = 0.0f;
        v2f b; b[0] = 1.0f; b[1] = 1.0f;
        v8f c = {};
        c = __builtin_amdgcn_wmma_f32_16x16x4_f32(
                /*neg_a=*/false, a, /*neg_b=*/false, b,
                /*c_mod=*/(short)0, c, /*reuse_a=*/false, /*reuse_b=*/false);

        // Lane L holds D rows M = {0..7}+8*(L>=16) at column N=L%16.
        float t = c[0] + c[1] + c[2] + c[3] + c[4] + c[5] + c[6] + c[7];
        t += __shfl_xor(t, 16, 32);   // add rows 8..15 half
        if (threadIdx.x == 0)
            out[0] = sqrtf(t / (float)n);
    }
}

extern "C" void kernel_launch(void* const* d_in, const int* in_sizes, int n_in,
                              void* d_out, int out_size, void* d_ws, size_t ws_size,
                              hipStream_t stream) {
    const float* in  = (const float*)d_in[0];   // inputs  [N,1] f32
    const float* tgt = (const float*)d_in[1];   // targets [N,1] f32
    float* out       = (float*)d_out;           // scalar f32
    float* partial   = (float*)d_ws;            // NBLOCKS floats of scratch
    const long long n = (long long)in_sizes[0];

    rmse_partial_kernel<<<NBLOCKS, NTHREADS, 0, stream>>>(in, tgt, partial, n);
    rmse_final_kernel<<<1, NTHREADS, 0, stream>>>(partial, out, n);
}